// UnfoldMatmul_82368882803237
// MI455X (gfx1250) — compile-verified
//
#include <hip/hip_runtime.h>

// out[b,h,t,f] = sum_{r=0..64} a[b,h,t,r] * v[b,h, t+r-32, f]   (zero-padded in t)
// Banded matmul per (b,h): out = M @ V, M[t,t'] = a[t, t'-t+32], band width 65.
// Each wave: 16(T) x 64(F) output tile = 20 K-steps of V_WMMA_F32_16X16X4_F32.
// LDS: 16x80 zero-padded "staircase" A tile + pair-interleaved 128-row V window
// so every A and B fragment is a single 8B-aligned ds_load_b64 (no VGPR shuffles).

typedef __attribute__((ext_vector_type(2))) float v2f;
typedef __attribute__((ext_vector_type(8))) float v8f;

#define T_DIM   4096
#define F_DIM   64
#define R_DIM   65
#define PAD     32          // R/2
#define TBLK    64          // t-rows per workgroup (4 waves x 16)
#define KW      80          // K-window per 16-row tile: 16 + 64
#define VROWS   128         // TBLK + 2*PAD v rows staged
#define VPSTR   128         // floats per interleaved row-pair (64 cols x 2)
#define NTHREADS 128

__global__ __launch_bounds__(NTHREADS)
void unfold_wmma_kernel(const float* __restrict__ a,
                        const float* __restrict__ v,
                        float* __restrict__ out) {
  __shared__ float sA[TBLK * KW];              // staircase A: 20480 B
  __shared__ float sV[(VROWS / 2) * VPSTR];    // pair-interleaved V: 32768 B
  // sV layout: element (k,f) at sV[(k>>1)*VPSTR + 2*f + (k&1)]
  //  -> B fragment {B[k,n], B[k+1,n]} (k even) is one contiguous, 8B-aligned b64.

  const int tid  = threadIdx.x;
  const int bh   = blockIdx.y;
  const int wgT0 = blockIdx.x * TBLK;

  // ---- zero the A staircase (borders must be 0) ----
  for (int e = tid; e < TBLK * KW; e += NTHREADS)
    sA[e] = 0.0f;

  // ---- stage v rows (zero-padded at T edges), float4 global reads, interleaved LDS writes ----
  const float* vBase = v + (size_t)bh * T_DIM * F_DIM;
  for (int idx = tid; idx < VROWS * (F_DIM / 4); idx += NTHREADS) {
    const int row = idx >> 4;           // 0..127
    const int c4  = (idx & 15) << 2;    // f offset 0..60
    const int gt  = wgT0 - PAD + row;
    float4 val = make_float4(0.f, 0.f, 0.f, 0.f);
    if (gt >= 0 && gt < T_DIM)
      val = *(const float4*)(vBase + (size_t)gt * F_DIM + c4);
    float* dst = &sV[(row >> 1) * VPSTR + (row & 1) + 2 * c4];
    dst[0] = val.x; dst[2] = val.y; dst[4] = val.z; dst[6] = val.w;
  }

  __syncthreads();   // sA fully zeroed before staircase scatter

  // ---- scatter a into the staircase: sA[row][ (row%16) + r ] = a[wgT0+row, r] ----
  // a block for this WG is a single contiguous run of TBLK*R floats -> coalesced reads.
  const float* aBlk = a + ((size_t)bh * T_DIM + wgT0) * R_DIM;
  for (int e = tid; e < TBLK * R_DIM; e += NTHREADS) {
    const int row = e / R_DIM;
    const int r   = e - row * R_DIM;
    sA[row * KW + (row & 15) + r] = aBlk[e];
  }

  __syncthreads();

  // ---- per-wave WMMA compute: wave w -> t-tile [wgT0+16w, +16), full F=64 ----
  const int lane = tid & 31;
  const int w    = tid >> 5;
  const int i    = lane & 15;     // M row (A), N col (B/C/D)
  const int g    = lane >> 4;     // lane-group: K split / M split

  v8f acc0 = {}, acc1 = {}, acc2 = {}, acc3 = {};

  // A 16x4 layout: lane (i,g), VGPR j holds A[i, k0 + j + 2g]  -> one b64
  const float* aRow = &sA[(16 * w + i) * KW + 2 * g];
  // B 4x16 layout: lane (n=i,g), VGPR j holds B[k0 + j + 2g, n] -> one b64
  // rows k0+2g, k0+1+2g share interleave-pair (8w + k0/2 + g)
  const float* vRow = &sV[(8 * w + g) * VPSTR + 2 * i];

  for (int k0 = 0; k0 < KW; k0 += 4) {
    const v2f af = *(const v2f*)(aRow + k0);
    const float* vp = vRow + k0 * (VPSTR / 2);      // advance 2 pairs per K-step
    const v2f b0 = *(const v2f*)(vp + 0);           // fc = 0
    const v2f b1 = *(const v2f*)(vp + 32);          // fc = 16
    const v2f b2 = *(const v2f*)(vp + 64);          // fc = 32
    const v2f b3 = *(const v2f*)(vp + 96);          // fc = 48
    acc0 = __builtin_amdgcn_wmma_f32_16x16x4_f32(false, af, false, b0, (short)0, acc0, false, false);
    acc1 = __builtin_amdgcn_wmma_f32_16x16x4_f32(false, af, false, b1, (short)0, acc1, false, false);
    acc2 = __builtin_amdgcn_wmma_f32_16x16x4_f32(false, af, false, b2, (short)0, acc2, false, false);
    acc3 = __builtin_amdgcn_wmma_f32_16x16x4_f32(false, af, false, b3, (short)0, acc3, false, false);
  }

  // ---- store: D 16x16 f32 layout -> VGPR p: M = p + 8g, N = i ----
  float* oBase = out + ((size_t)bh * T_DIM + wgT0 + 16 * w) * F_DIM;
  #pragma unroll
  for (int p = 0; p < 8; ++p) {
    float* orow = oBase + (size_t)(p + 8 * g) * F_DIM + i;
    orow[0]  = acc0[p];
    orow[16] = acc1[p];
    orow[32] = acc2[p];
    orow[48] = acc3[p];
  }
}

extern "C" void kernel_launch(void* const* d_in, const int* in_sizes, int n_in,
                              void* d_out, int out_size, void* d_ws, size_t ws_size,
                              hipStream_t stream) {
  const float* a   = (const float*)d_in[0];
  const float* v   = (const float*)d_in[1];
  float*       out = (float*)d_out;

  const int BH = in_sizes[1] / (T_DIM * F_DIM);   // B*H = 128
  dim3 grid(T_DIM / TBLK, BH);
  unfold_wmma_kernel<<<grid, NTHREADS, 0, stream>>>(a, v, out);
}